// Downsampler_9740985828153
// MI455X (gfx1250) — compile-verified
//
#include <hip/hip_runtime.h>

// Downsampler: 9x9 Gaussian, stride 4, pad 4, fp32.
// In:  (4,64,512,512) f32   Out: (4,64,128,128) f32
// Memory-bound: ~285 MB HBM traffic -> ~12us floor at 23.3 TB/s.
// Strategy: fused tiled conv, read HBM once via CDNA5 async global->LDS b128
// copies (GLOBAL_LOAD_ASYNC_TO_LDS_B128 + s_wait_asynccnt), prefetch next
// tile (global_prefetch_b8), 27 ds ops per output (b128 tap loads, 16B
// aligned, conflict-free). No WMMA: 0.68 GFLOP of f32 FMAs is far below the
// HBM floor, so matrix ops would only add layout-shuffle overhead.

#define BC    256          // B*C channel images
#define HW    512          // input H == W
#define OWH   128          // output H == W
#define KS    9
#define STRD  4
#define PADI  4
#define TILE  4                          // output rows per LDS tile
#define ROWS  (TILE * STRD + KS - STRD)  // 21 input rows per tile
#define LDSW  524                        // floats/LDS row: 4 halo + 512 + 8 pad (16B-multiple)
#define TPG   4                          // tiles processed per block
#define GPB   (OWH / (TILE * TPG))       // 8 tile-groups per channel image

typedef float f4 __attribute__((ext_vector_type(4)));
typedef int v4i __attribute__((vector_size(16)));   // matches builtin param type

// ---- CDNA5 async global->LDS copy (16B), with safe fallback ----------------
__device__ __forceinline__ void g2l_b128(const float* g, float* l) {
#if defined(__HIP_DEVICE_COMPILE__) && \
    __has_builtin(__builtin_amdgcn_global_load_async_to_lds_b128)
    // Param 0: v4i addrspace(1)* (global src); param 1: v4i addrspace(3)* (LDS
    // dst; flat addr low 32 bits == wave-relative LDS byte offset on AMDGCN).
    __builtin_amdgcn_global_load_async_to_lds_b128(
        (__attribute__((address_space(1))) v4i*)(unsigned long long)g,
        (__attribute__((address_space(3))) v4i*)(unsigned)(unsigned long long)l,
        0, 0);
#else
    *(f4*)l = *(const f4*)g;
#endif
}

__device__ __forceinline__ void wait_async_copies() {
#if defined(__HIP_DEVICE_COMPILE__) && \
    __has_builtin(__builtin_amdgcn_global_load_async_to_lds_b128)
#if __has_builtin(__builtin_amdgcn_s_wait_asynccnt)
    __builtin_amdgcn_s_wait_asynccnt(0);
#else
    asm volatile("s_wait_asynccnt 0" ::: "memory");
#endif
#endif
}

__global__ __launch_bounds__(256)
void Downsampler_9740985828153_kernel(const float* __restrict__ in,
                                      const float* __restrict__ kw,
                                      float* __restrict__ out) {
    __shared__ __align__(16) float sm[ROWS * LDSW];  // ~43 KB input tile
    __shared__ __align__(16) float wl[KS * KS];      // 9x9 weights

    const int tid = threadIdx.x;

    // ---- weights -> LDS via async b128 (first instructions of the kernel) ----
    // 20 x 16B covers w[0..79]; w[80] copied scalar (never read past 81 floats).
    if (tid < 20)       g2l_b128(kw + tid * 4, &wl[tid * 4]);
    else if (tid == 20) wl[80] = kw[80];

    const int bc  = blockIdx.x / GPB;   // channel image 0..255
    const int tg  = blockIdx.x % GPB;   // tile group 0..7
    const float* __restrict__ chan = in + (size_t)bc * (HW * HW);

    for (int t = 0; t < TPG; ++t) {
        const int oy0 = (tg * TPG + t) * TILE;  // first output row of tile
        const int iy0 = oy0 * STRD - PADI;      // first input row of tile

        // ---- stage 21x512 interior floats into LDS (async b128) ----
        for (int idx = tid; idx < ROWS * (HW / 4); idx += 256) {
            const int r  = idx >> 7;            // HW/4 == 128
            const int c  = (idx & 127) * 4;     // float column
            const int iy = iy0 + r;
            float* dst = &sm[r * LDSW + PADI + c];
            if ((unsigned)iy < (unsigned)HW) {
                g2l_b128(chan + (size_t)iy * HW + c, dst);
            } else {
                f4 z = {0.f, 0.f, 0.f, 0.f};
                *(f4*)dst = z;                  // top/bottom zero padding
            }
        }
        // ---- zero the x halos (cols 0..3 and 516..519) ----
        if (tid < ROWS) {
            f4 z = {0.f, 0.f, 0.f, 0.f};
            *(f4*)&sm[tid * LDSW + 0]         = z;
            *(f4*)&sm[tid * LDSW + PADI + HW] = z;
        }
        // ---- prefetch next tile's rows (gfx1250 global_prefetch_b8) ----
        if (t + 1 < TPG) {
            const int niy0 = iy0 + TILE * STRD;
            const int pr = tid >> 3;            // 21 rows x 8 segments of 256B
            const int ps = tid & 7;
            if (pr < ROWS) {
                const int iy = niy0 + pr;
                if ((unsigned)iy < (unsigned)HW)
                    __builtin_prefetch(chan + (size_t)iy * HW + ps * 64, 0, 1);
            }
        }
        wait_async_copies();
        __syncthreads();

        // ---- compute: 2 outputs per thread, 9 rows x (b128,b128,b32) taps ----
#pragma unroll
        for (int p = 0; p < 2; ++p) {
            const int lin = tid + p * 256;
            const int oyl = lin >> 7;           // 0..3  (local output row)
            const int ox  = lin & 127;          // output column
            float acc = 0.f;
#pragma unroll
            for (int j = 0; j < KS; ++j) {
                // taps: input x = ox*4 - 4 + i  ->  LDS col = ox*4 + i (16B aligned)
                const float* row = &sm[(oyl * STRD + j) * LDSW + ox * 4];
                f4 a = *(const f4*)(row);
                f4 b = *(const f4*)(row + 4);
                float c8 = row[8];
                const float* w = &wl[j * KS];
                acc = fmaf(w[0], a.x, acc);
                acc = fmaf(w[1], a.y, acc);
                acc = fmaf(w[2], a.z, acc);
                acc = fmaf(w[3], a.w, acc);
                acc = fmaf(w[4], b.x, acc);
                acc = fmaf(w[5], b.y, acc);
                acc = fmaf(w[6], b.z, acc);
                acc = fmaf(w[7], b.w, acc);
                acc = fmaf(w[8], c8, acc);
            }
            out[((size_t)bc * OWH + (oy0 + oyl)) * OWH + ox] = acc;
        }
        __syncthreads();  // LDS reused by next tile
    }
}

extern "C" void kernel_launch(void* const* d_in, const int* in_sizes, int n_in,
                              void* d_out, int out_size, void* d_ws, size_t ws_size,
                              hipStream_t stream) {
    (void)in_sizes; (void)n_in; (void)out_size; (void)d_ws; (void)ws_size;
    const float* in  = (const float*)d_in[0];   // 4*64*512*512 f32
    const float* kw  = (const float*)d_in[1];   // 81 f32 (9x9 kernel)
    float* out = (float*)d_out;                 // 4*64*128*128 f32
    dim3 grid(BC * GPB);                        // 2048 blocks
    Downsampler_9740985828153_kernel<<<grid, 256, 0, stream>>>(in, kw, out);
}